// SCAESuite_49546742726742
// MI455X (gfx1250) — compile-verified
//
#include <hip/hip_runtime.h>

// Problem constants (fixed by the reference).
constexpr int BATCH = 2;
constexpr int SEQ   = 512;
constexpr int F     = 16384;   // n_features
constexpr int D     = 768;     // d_model
constexpr int M     = 262144;  // n_connections (i_indices sorted)
constexpr int NROW  = BATCH * SEQ;        // 1024 "rows" of up_facts / out
constexpr int NGROUP = F / 16;            // 1024 feature groups of 16
constexpr int NTILE  = NROW / 16;         // 64 row tiles of 16

typedef __attribute__((ext_vector_type(2))) float v2f;
typedef __attribute__((ext_vector_type(8))) float v8f;

// ---------------------------------------------------------------------------
// Phase 1: values[m] = dot(down_encoder[i_m, :], up_decoder[:, j_m])
// One wave computes 16 values via V_WMMA_F32_16X16X4_F32 accumulated over
// K = 768 in steps of 4; we keep only the diagonal of the 16x16 result.
// Two independent accumulators (even/odd K-chunks) break the WMMA->WMMA
// C-chain so the XDL pipe stays full; summed once at the end.
//
// fp32 A 16x4 layout: lanes 0-15 hold M=lane, K={0,1} in VGPR{0,1};
//                     lanes 16-31 hold M=lane-16, K={2,3}.
// fp32 B 4x16 layout (symmetric): lanes 0-15 N=lane K={0,1}; lanes 16-31 K={2,3}.
// fp32 C/D 16x16: lane l, VGPR v -> (M = v + (l>=16)*8, N = l&15).
// ---------------------------------------------------------------------------
__global__ __launch_bounds__(256)
void sdd_values_wmma(const float* __restrict__ DE,   // [F, D] row-major
                     const float* __restrict__ UD,   // [D, F] row-major
                     const int*   __restrict__ iIdx, // [M] sorted
                     const int*   __restrict__ jIdx, // [M]
                     float*       __restrict__ values) {
  const int wave = (blockIdx.x * blockDim.x + threadIdx.x) >> 5;
  const int lane = threadIdx.x & 31;
  const int m0   = wave << 4;                 // 16 connections per wave
  const int half = lane >> 4;                 // 0: K={0,1}; 1: K={2,3}
  const int sub  = lane & 15;

  const int rowA = iIdx[m0 + sub];            // A-matrix row M=sub
  const int colB = jIdx[m0 + sub];            // B-matrix col N=sub
  const float* Arow = DE + rowA * D;          // contiguous row
  const float* Bcol = UD + colB;              // stride-F column (L2 resident)

  v8f c0 = {0.f, 0.f, 0.f, 0.f, 0.f, 0.f, 0.f, 0.f};
  v8f c1 = {0.f, 0.f, 0.f, 0.f, 0.f, 0.f, 0.f, 0.f};
  for (int k = 0; k < D; k += 8) {
    const int ka = k + half * 2;
    const int kb = ka + 4;
    v2f a0, b0, a1, b1;
    a0.x = Arow[ka];
    a0.y = Arow[ka + 1];
    b0.x = Bcol[ka * F];
    b0.y = Bcol[(ka + 1) * F];
    a1.x = Arow[kb];
    a1.y = Arow[kb + 1];
    b1.x = Bcol[kb * F];
    b1.y = Bcol[(kb + 1) * F];
    c0 = __builtin_amdgcn_wmma_f32_16x16x4_f32(false, a0, false, b0,
                                               (short)0, c0, false, false);
    c1 = __builtin_amdgcn_wmma_f32_16x16x4_f32(false, a1, false, b1,
                                               (short)0, c1, false, false);
  }
  const v8f c = c0 + c1;

  // Diagonal (t,t): t<8  -> lane t,    VGPR t
  //                 t>=8 -> lane t+16, VGPR t-8
  if (lane < 8 || lane >= 24) {
    const int s = (lane < 8) ? lane : (lane - 24);   // VGPR index 0..7
    const int t = (lane < 8) ? lane : (lane - 16);   // diagonal index 0..15
    float r = (s < 4) ? ((s < 2) ? ((s == 0) ? c[0] : c[1])
                                 : ((s == 2) ? c[2] : c[3]))
                      : ((s < 6) ? ((s == 4) ? c[4] : c[5])
                                 : ((s == 6) ? c[6] : c[7]));
    values[m0 + t] = r;
  }
}

__device__ __forceinline__ int lower_bound_i(const int* __restrict__ a, int n, int key) {
  int lo = 0, hi = n;
  while (lo < hi) {
    int mid = (lo + hi) >> 1;
    if (a[mid] < key) lo = mid + 1; else hi = mid;
  }
  return lo;
}

// ---------------------------------------------------------------------------
// Phase 2: out[n, i] = sum_{m : i_m == i} values[m] * up[n, j_m]
// Sorted i => feature group g (features 16g..16g+15) owns a CONTIGUOUS m-range
// found by binary search (no atomics, no scatter). Each wave computes one
// 16(row) x 16(feature) output tile: accumulate 4-connection chunks as
//   A(16 x 4)  = gathered up_facts columns j_m   (L2-resident gathers)
//   B(4 x 16)  = one-hot: B[k][col] = values[m_k] iff i_{m_k} == 16g+col
// Manually unrolled 2 chunks/iteration into 2 accumulators; per-slot clamping
// (B=0 past segment end) makes the tail safe without an epilogue.
// Every tile is written (zeros for empty groups) -> full output init.
// ---------------------------------------------------------------------------
__global__ __launch_bounds__(256)
void spmm_wmma(const float* __restrict__ up,     // [NROW, F]
               const int*   __restrict__ iIdx,   // [M] sorted
               const int*   __restrict__ jIdx,   // [M]
               const float* __restrict__ values, // [M]
               float*       __restrict__ out) {  // [NROW, F]
  const int task = (blockIdx.x * blockDim.x + threadIdx.x) >> 5;
  const int lane = threadIdx.x & 31;
  const int g = task >> 6;                  // feature group (waves in a block
  const int t = task & 63;                  // share g -> index loads hit L0)
  const int fbase = g << 4;

  const int lo = lower_bound_i(iIdx, M, fbase);
  const int hi = lower_bound_i(iIdx, M, fbase + 16);

  const int half = lane >> 4;
  const int sub  = lane & 15;
  const int fcol = fbase + sub;
  const float* uprow = up + (t * 16 + sub) * F;   // A row M=sub

  v8f c0 = {0.f, 0.f, 0.f, 0.f, 0.f, 0.f, 0.f, 0.f};
  v8f c1 = {0.f, 0.f, 0.f, 0.f, 0.f, 0.f, 0.f, 0.f};
  for (int m = lo; m < hi; m += 8) {
    // ---- chunk 0: connections m .. m+3 ----
    {
      const int k0 = m + half * 2;
      const int k1 = k0 + 1;
      const bool ok0 = (k0 < hi), ok1 = (k1 < hi);
      const int  s0  = ok0 ? k0 : lo;
      const int  s1  = ok1 ? k1 : lo;
      v2f a, b;
      b.x = (ok0 && iIdx[s0] == fcol) ? values[s0] : 0.f;
      b.y = (ok1 && iIdx[s1] == fcol) ? values[s1] : 0.f;
      a.x = uprow[jIdx[s0]];
      a.y = uprow[jIdx[s1]];
      c0 = __builtin_amdgcn_wmma_f32_16x16x4_f32(false, a, false, b,
                                                 (short)0, c0, false, false);
    }
    // ---- chunk 1: connections m+4 .. m+7 ----
    {
      const int k0 = m + 4 + half * 2;
      const int k1 = k0 + 1;
      const bool ok0 = (k0 < hi), ok1 = (k1 < hi);
      const int  s0  = ok0 ? k0 : lo;
      const int  s1  = ok1 ? k1 : lo;
      v2f a, b;
      b.x = (ok0 && iIdx[s0] == fcol) ? values[s0] : 0.f;
      b.y = (ok1 && iIdx[s1] == fcol) ? values[s1] : 0.f;
      a.x = uprow[jIdx[s0]];
      a.y = uprow[jIdx[s1]];
      c1 = __builtin_amdgcn_wmma_f32_16x16x4_f32(false, a, false, b,
                                                 (short)0, c1, false, false);
    }
  }
  const v8f c = c0 + c1;

  // Store 16x16 tile: lane l -> col = l&15; VGPR v -> row = 16t + v + 8*half
  float* otile = out + (t * 16 + half * 8) * F + fbase + sub;
#pragma unroll
  for (int v = 0; v < 8; ++v) {
    otile[v * F] = c[v];
  }
}

extern "C" void kernel_launch(void* const* d_in, const int* in_sizes, int n_in,
                              void* d_out, int out_size, void* d_ws, size_t ws_size,
                              hipStream_t stream) {
  const float* up_facts     = (const float*)d_in[0];  // [B, S, F]
  const float* down_encoder = (const float*)d_in[1];  // [F, D]
  const float* up_decoder   = (const float*)d_in[2];  // [D, F]
  const int*   i_indices    = (const int*)d_in[3];    // [M] sorted
  const int*   j_indices    = (const int*)d_in[4];    // [M]
  float*       out          = (float*)d_out;          // [B, S, F]
  float*       values       = (float*)d_ws;           // [M] scratch (1 MB)

  // Phase 1: M/16 = 16384 waves, 8 waves per 256-thread block.
  sdd_values_wmma<<<(M / 16) / 8, 256, 0, stream>>>(down_encoder, up_decoder,
                                                    i_indices, j_indices, values);

  // Phase 2: NGROUP*NTILE = 65536 wave-tiles, 8 per block.
  spmm_wmma<<<(NGROUP * NTILE) / 8, 256, 0, stream>>>(up_facts, i_indices,
                                                      j_indices, values, out);
}